// GATLayer_22153441313024
// MI455X (gfx1250) — compile-verified
//
#include <hip/hip_runtime.h>
#include <hip/hip_bf16.h>
#include <math.h>

#define N_NODES 100000
#define NUM_EDGES 1000000
#define IN_F 128
#define OUT_F 64
#define ALPHA 0.2f

typedef __attribute__((ext_vector_type(2))) float v2f;
typedef __attribute__((ext_vector_type(8))) float v8f;

// ---------------------------------------------------------------------------
// float atomic max via signed/unsigned integer atomics (handles mixed signs,
// buffer initialized to -inf bit pattern 0xFF800000)
// ---------------------------------------------------------------------------
__device__ __forceinline__ void atomicMaxFloat(float* addr, float val) {
    if (val >= 0.0f) {
        atomicMax((int*)addr, __float_as_int(val));
    } else {
        atomicMin((unsigned int*)addr, __float_as_uint(val));
    }
}

// ---------------------------------------------------------------------------
// Kernel 0: initialize out (N*64 zeros), maxe (-inf), sume (0)
// ---------------------------------------------------------------------------
__global__ void gat_init(float* __restrict__ out, float* __restrict__ maxe,
                         float* __restrict__ sume) {
    int i = blockIdx.x * blockDim.x + threadIdx.x;
    if (i < N_NODES * OUT_F) out[i] = 0.0f;
    if (i < N_NODES) {
        maxe[i] = __int_as_float(0xff800000);  // -inf
        sume[i] = 0.0f;
    }
}

// ---------------------------------------------------------------------------
// Kernel 1: h = x @ W^T  via V_WMMA_F32_16X16X4_F32
//   block = 128 threads = 4 waves; wave w computes the 16x16 tile at
//   rows [16*blockIdx.x, +16), cols [16*w, +16).  K = 128 -> 32 WMMA ops.
//
// fp32 A (16x4) layout:  lane L: m=L&15, kpair=(L>>4)*2 -> {K=kp, K=kp+1}
// fp32 B (4x16) layout:  lane L: n=L&15, kpair=(L>>4)*2 -> {K=kp, K=kp+1}
// fp32 C/D (16x16):      VGPR v: row = v + 8*(L>>4), col = L&15
// ---------------------------------------------------------------------------
__global__ void gat_gemm(const float* __restrict__ x, const float* __restrict__ W,
                         float* __restrict__ h) {
    const int wave = threadIdx.x >> 5;          // 0..3 -> column tile
    const int lane = threadIdx.x & 31;
    const int row0 = blockIdx.x * 16;
    const int col0 = wave * 16;
    const int mn = lane & 15;                    // m for A, n for B
    const int kb = (lane >> 4) * 2;              // 0 or 2

    const float* __restrict__ xrow = x + (size_t)(row0 + mn) * IN_F;
    const float* __restrict__ wrow = W + (size_t)(col0 + mn) * IN_F; // B[k][n]=W[n][k]

    v8f c = {};
#pragma unroll
    for (int k = 0; k < IN_F; k += 4) {
        v2f a, b;
        a.x = xrow[k + kb];
        a.y = xrow[k + kb + 1];
        b.x = wrow[k + kb];
        b.y = wrow[k + kb + 1];
        c = __builtin_amdgcn_wmma_f32_16x16x4_f32(
                /*neg_a=*/false, a, /*neg_b=*/false, b,
                /*c_mod=*/(short)0, c, /*reuse_a=*/false, /*reuse_b=*/false);
    }

    const int n = lane & 15;
    const int mbase = (lane >> 4) * 8;
#pragma unroll
    for (int v = 0; v < 8; ++v) {
        h[(size_t)(row0 + mbase + v) * OUT_F + col0 + n] = c[v];
    }
}

// ---------------------------------------------------------------------------
// Kernel 2: per-node attention coefficients  a_src[n]=h[n]·a[0:64],
//           a_dst[n]=h[n]·a[64:128]
// ---------------------------------------------------------------------------
__global__ void gat_node_coef(const float* __restrict__ h, const float* __restrict__ a,
                              float* __restrict__ asrc, float* __restrict__ adst) {
    int n = blockIdx.x * blockDim.x + threadIdx.x;
    if (n >= N_NODES) return;
    const float4* hr = (const float4*)(h + (size_t)n * OUT_F);
    const float4* a0 = (const float4*)(a);
    const float4* a1 = (const float4*)(a + OUT_F);
    float s0 = 0.0f, s1 = 0.0f;
#pragma unroll
    for (int f = 0; f < OUT_F / 4; ++f) {
        float4 hv = hr[f], av = a0[f], bv = a1[f];
        s0 += hv.x * av.x + hv.y * av.y + hv.z * av.z + hv.w * av.w;
        s1 += hv.x * bv.x + hv.y * bv.y + hv.z * bv.z + hv.w * bv.w;
    }
    asrc[n] = s0;
    adst[n] = s1;
}

// ---------------------------------------------------------------------------
// Kernel 3: e = LeakyReLU(asrc[src]+adst[dst]); segment max over dst
// ---------------------------------------------------------------------------
__global__ void gat_edge_logits(const long long* __restrict__ ei,
                                const float* __restrict__ asrc,
                                const float* __restrict__ adst,
                                float* __restrict__ ebuf, float* __restrict__ maxe) {
    int e = blockIdx.x * blockDim.x + threadIdx.x;
    if (e >= NUM_EDGES) return;
    int s = (int)ei[e];
    int d = (int)ei[NUM_EDGES + e];
    float v = asrc[s] + adst[d];
    v = (v > 0.0f) ? v : ALPHA * v;
    ebuf[e] = v;
    atomicMaxFloat(&maxe[d], v);
}

// ---------------------------------------------------------------------------
// Kernel 4: exp_e = exp(e - maxe[dst]); segment sum over dst
// ---------------------------------------------------------------------------
__global__ void gat_edge_exp(const long long* __restrict__ ei,
                             float* __restrict__ ebuf,
                             const float* __restrict__ maxe,
                             float* __restrict__ sume) {
    int e = blockIdx.x * blockDim.x + threadIdx.x;
    if (e >= NUM_EDGES) return;
    int d = (int)ei[NUM_EDGES + e];
    float v = __expf(ebuf[e] - maxe[d]);
    ebuf[e] = v;
    atomicAdd(&sume[d], v);
}

// ---------------------------------------------------------------------------
// Kernel 5: out[dst] += h[src] * (exp_e / (sume[dst] + 1e-16))
//   64 consecutive threads handle the 64 features of one edge.
//   h and out both fit in the 192 MB L2, so gathers + atomics stay on-chip.
// ---------------------------------------------------------------------------
__global__ void gat_scatter(const long long* __restrict__ ei,
                            const float* __restrict__ ebuf,
                            const float* __restrict__ sume,
                            const float* __restrict__ h, float* __restrict__ out) {
    size_t idx = (size_t)blockIdx.x * blockDim.x + threadIdx.x;
    size_t e = idx >> 6;
    int f = (int)(idx & 63);
    if (e >= NUM_EDGES) return;
    int s = (int)ei[e];
    int d = (int)ei[NUM_EDGES + e];
    float attn = ebuf[e] / (sume[d] + 1e-16f);
    atomicAdd(&out[(size_t)d * OUT_F + f], h[(size_t)s * OUT_F + f] * attn);
}

// ---------------------------------------------------------------------------
// Kernel 6: ELU in place
// ---------------------------------------------------------------------------
__global__ void gat_elu(float* __restrict__ out) {
    int i = blockIdx.x * blockDim.x + threadIdx.x;
    if (i >= N_NODES * OUT_F) return;
    float v = out[i];
    out[i] = (v > 0.0f) ? v : (__expf(v) - 1.0f);
}

// ---------------------------------------------------------------------------
extern "C" void kernel_launch(void* const* d_in, const int* in_sizes, int n_in,
                              void* d_out, int out_size, void* d_ws, size_t ws_size,
                              hipStream_t stream) {
    const float*     x  = (const float*)d_in[0];
    const long long* ei = (const long long*)d_in[1];  // int64 [2, E]
    const float*     W  = (const float*)d_in[2];
    const float*     a  = (const float*)d_in[3];
    float* out = (float*)d_out;

    // workspace layout
    float* h    = (float*)d_ws;                          // N * 64
    float* asrc = h + (size_t)N_NODES * OUT_F;           // N
    float* adst = asrc + N_NODES;                        // N
    float* maxe = adst + N_NODES;                        // N
    float* sume = maxe + N_NODES;                        // N
    float* ebuf = sume + N_NODES;                        // E

    const int B = 256;

    gat_init<<<(N_NODES * OUT_F + B - 1) / B, B, 0, stream>>>(out, maxe, sume);

    gat_gemm<<<N_NODES / 16, 128, 0, stream>>>(x, W, h);

    gat_node_coef<<<(N_NODES + B - 1) / B, B, 0, stream>>>(h, a, asrc, adst);

    gat_edge_logits<<<(NUM_EDGES + B - 1) / B, B, 0, stream>>>(ei, asrc, adst, ebuf, maxe);

    gat_edge_exp<<<(NUM_EDGES + B - 1) / B, B, 0, stream>>>(ei, ebuf, maxe, sume);

    gat_scatter<<<(int)(((size_t)NUM_EDGES * OUT_F) / B), B, 0, stream>>>(ei, ebuf, sume, h, out);

    gat_elu<<<(N_NODES * OUT_F + B - 1) / B, B, 0, stream>>>(out);
}